// PositionalTopDownHTMM_39762807227043
// MI455X (gfx1250) — compile-verified
//
#include <hip/hip_runtime.h>
#include <math.h>

// CDNA5 wave32 WMMA types
typedef float v2f __attribute__((ext_vector_type(2)));
typedef float v4f __attribute__((ext_vector_type(4)));
typedef float v8f __attribute__((ext_vector_type(8)));

// Fast natural log: raw v_log_f32 (log2) * ln2. Inputs here are sums of
// softmax-probability products, always normal-range => no denorm prescale needed.
__device__ __forceinline__ float fast_log(float x) {
    return __builtin_amdgcn_logf(x) * 0.69314718055994530942f;
}

// Problem constants
// B_TREES=32, BR=4, DEPTH=7, C=8, G=16, M=256
// sm_A index: ((i*8 + j)*4 + p)*16 + g  = i*512 + j*64 + p*16 + g
// sm_B index: (c*256 + m)*16 + g        = c*4096 + m*16 + g

__global__ void zero_out_kernel(float* out) {
    int t = threadIdx.x;
    if (t < 512) out[t] = 0.0f;
}

// Softmaxes: sm_A over i (512 groups), sm_B over m (128 groups), sm_Pi over c (16 groups).
__global__ void softmax_kernel(const float* __restrict__ A,
                               const float* __restrict__ B,
                               const float* __restrict__ Pi,
                               float* __restrict__ smA,
                               float* __restrict__ smB,
                               float* __restrict__ pc0) {
    int gid = blockIdx.x * blockDim.x + threadIdx.x;
    if (gid < 512) {
        // group = (j,p,g): base offset = gid, stride over i = 512
        float mx = -INFINITY;
        #pragma unroll
        for (int i = 0; i < 8; ++i) mx = fmaxf(mx, A[i * 512 + gid]);
        float e[8], s = 0.0f;
        #pragma unroll
        for (int i = 0; i < 8; ++i) { e[i] = __expf(A[i * 512 + gid] - mx); s += e[i]; }
        float inv = __builtin_amdgcn_rcpf(s);
        #pragma unroll
        for (int i = 0; i < 8; ++i) smA[i * 512 + gid] = e[i] * inv;
    } else if (gid < 640) {
        int gb = gid - 512;                // (c,g)
        int c = gb >> 4, g = gb & 15;
        int base = c * 4096 + g;           // stride over m = 16
        float mx = -INFINITY;
        for (int m = 0; m < 256; ++m) mx = fmaxf(mx, B[base + m * 16]);
        float s = 0.0f;
        for (int m = 0; m < 256; ++m) s += __expf(B[base + m * 16] - mx);
        float inv = __builtin_amdgcn_rcpf(s);
        for (int m = 0; m < 256; ++m) smB[base + m * 16] = __expf(B[base + m * 16] - mx) * inv;
    } else if (gid < 656) {
        int g = gid - 640;
        float mx = -INFINITY;
        #pragma unroll
        for (int c = 0; c < 8; ++c) mx = fmaxf(mx, Pi[c * 16 + g]);
        float e[8], s = 0.0f;
        #pragma unroll
        for (int c = 0; c < 8; ++c) { e[c] = __expf(Pi[c * 16 + g] - mx); s += e[c]; }
        float inv = __builtin_amdgcn_rcpf(s);
        #pragma unroll
        for (int c = 0; c < 8; ++c) pc0[c * 16 + g] = e[c] * inv;
    }
}

// Canonical priors, one level: pc[poff+m][i][g] = sum_j smA[i,j,m%4,g] * pc[poffp+m/4][j][g]
__global__ void prior_kernel(const float* __restrict__ smA,
                             float* __restrict__ pc,
                             int level, int poffp, int poff) {
    int tid = blockIdx.x * blockDim.x + threadIdx.x;
    int total = (1 << (2 * level)) * 128;
    if (tid >= total) return;
    int g = tid & 15;
    int i = (tid >> 4) & 7;
    int m = tid >> 7;
    int p = m & 3;
    const float* par = pc + (size_t)(poffp + (m >> 2)) * 128;
    float s = 0.0f;
    #pragma unroll
    for (int j = 0; j < 8; ++j)
        s += smA[i * 512 + j * 64 + p * 16 + g] * par[j * 16 + g];
    pc[(size_t)(poff + m) * 128 + i * 16 + g] = s;
}

// Upward pass for one level l (6..1). 128 threads = 4 waves; wave w handles the
// 16 nodes of the 64-node block with n%4==w.
//   prev = prior*b_em ; bpa = (l==6) ? prev : prev^2*msg_in ; nu = sum_c bpa
//   lognu -> per-tree LL accumulation
//   ratio = bpa/(nu*prior):
//     level 6: ratio[i] = bem[i] * (1/nu)            (prior cancels)
//     else:    ratio[i] = prev[i]*bem[i]*ms[i]*(1/nu)
//   beta_uv[j,g] = sum_i smA[i,j,w,g]*ratio[i,g]     (V_WMMA_F32_16X16X4_F32 x2, K=8)
//   msg_out[parent][j][g] = prod over 4 siblings of beta_uv
__global__ __launch_bounds__(128, 1)
void up_kernel(const float* __restrict__ smA,
               const float* __restrict__ smB,
               const float* __restrict__ pc,
               const int*   __restrict__ x,
               const float* __restrict__ msg_in,   // null for l==6
               float*       __restrict__ msg_out,
               float*       __restrict__ ll_out,
               int level, int poff, int xoff) {
    __shared__ float lds_buv[4][16][8][16];   // [wave][node-in-tile][j][g]
    __shared__ float lds_ll[16][16];          // [tree-slot][g]

    int tid = threadIdx.x;
    int w   = tid >> 5;        // wave index == position p
    int lam = tid & 31;
    int h   = lam >> 4;        // half (selects i-pairs per ISA A 16x4 layout)
    int t   = lam & 15;        // node index within wave tile; also B column j
    int base = blockIdx.x * 64;
    int n = base + 4 * t + w;  // this lane's node
    int lmask = (1 << (2 * level)) - 1;
    int m = n & lmask;         // within-tree canonical index
    int slot = (n >> (2 * level)) - (base >> (2 * level));

    if (tid < 128) {
        ((float*)lds_ll)[tid] = 0.0f;
        ((float*)lds_ll)[tid + 128] = 0.0f;
    }
    __syncthreads();

    int xi = x[xoff + n];
    const float* prn  = pc + (size_t)(poff + m) * 128;
    const float* bemp = smB + xi * 16;                 // + c*4096 + g
    const float* msp  = msg_in + (size_t)n * 128;      // only deref if level<6

    for (int g4 = 0; g4 < 16; g4 += 4) {
        // Per-c contributions: nu4 accumulates bpa, ru[c] = bpa/prior (ratio numerator)
        v4f nu4 = (v4f)0.0f;
        v4f ru[8];
        if (level == 6) {
            #pragma unroll
            for (int c = 0; c < 8; ++c) {
                v4f bem = *(const v4f*)(bemp + c * 4096 + g4);
                v4f pr  = *(const v4f*)(prn + c * 16 + g4);
                nu4 += pr * bem;
                ru[c] = bem;
            }
        } else {
            #pragma unroll
            for (int c = 0; c < 8; ++c) {
                v4f bem = *(const v4f*)(bemp + c * 4096 + g4);
                v4f pr  = *(const v4f*)(prn + c * 16 + g4);
                v4f ms  = *(const v4f*)(msp + c * 16 + g4);
                v4f prev = pr * bem;
                nu4 += prev * prev * ms;
                ru[c] = prev * bem * ms;
            }
        }
        // Select this lane's i-pairs once (2-way vector selects, h in {0,1}):
        // A 16x4 f32 layout: lane (h,t): A0 = ratio[t, 2h], ratio[t, 2h+1]; A1 = i+4.
        v4f rv0 = h ? ru[2] : ru[0];
        v4f rv1 = h ? ru[3] : ru[1];
        v4f rv2 = h ? ru[6] : ru[4];
        v4f rv3 = h ? ru[7] : ru[5];

        // B-matrix columns for this lane (col j = t), 4 g at a time
        // B 4x16 f32 layout: lane (h, col): vgpr0 = Ahat[2h, col], vgpr1 = Ahat[2h+1, col]
        v4f bv0, bv1, bv2, bv3;
        if (t < 8) {
            int jb = t * 64 + w * 16 + g4 + h * 2 * 512;  // i base = 2h
            bv0 = *(const v4f*)(smA + jb);                // i = 2h
            bv1 = *(const v4f*)(smA + jb + 512);          // i = 2h+1
            bv2 = *(const v4f*)(smA + jb + 4 * 512);      // i = 4+2h
            bv3 = *(const v4f*)(smA + jb + 5 * 512);      // i = 5+2h
        } else {
            bv0 = (v4f)0.0f; bv1 = (v4f)0.0f; bv2 = (v4f)0.0f; bv3 = (v4f)0.0f;
        }

        #pragma unroll
        for (int s = 0; s < 4; ++s) {
            int g = g4 + s;
            float nu = nu4[s];
            if (h == 0) {
                atomicAdd(&lds_ll[slot][g], fast_log(nu));
            }
            float inv = __builtin_amdgcn_rcpf(nu);

            v2f a0, a1;
            a0.x = rv0[s] * inv;
            a0.y = rv1[s] * inv;
            a1.x = rv2[s] * inv;
            a1.y = rv3[s] * inv;

            v2f b0, b1;
            b0.x = bv0[s]; b0.y = bv1[s];
            b1.x = bv2[s]; b1.y = bv3[s];

            v8f c0 = {};
            v8f d;
            d = __builtin_amdgcn_wmma_f32_16x16x4_f32(false, a0, false, b0, (short)0, c0, false, false);
            d = __builtin_amdgcn_wmma_f32_16x16x4_f32(false, a1, false, b1, (short)0, d,  false, false);

            // D layout: vgpr v, lane: row M = v + 8*h, col N = t. Keep j = t < 8.
            if (t < 8) {
                #pragma unroll
                for (int v = 0; v < 8; ++v)
                    lds_buv[w][v + 8 * h][t][g] = d[v];
            }
        }
    }
    __syncthreads();

    // Sibling product -> msg for parents: 16 parents x 8 j x 4 g-chunks = 512 float4 items
    #pragma unroll
    for (int k = 0; k < 4; ++k) {
        int item = tid + k * 128;
        int parL = item >> 5;          // 0..15
        int j    = (item >> 2) & 7;    // 0..7
        int g4   = (item & 3) * 4;     // 0,4,8,12
        v4f p0 = *(const v4f*)&lds_buv[0][parL][j][g4];
        v4f p1 = *(const v4f*)&lds_buv[1][parL][j][g4];
        v4f p2 = *(const v4f*)&lds_buv[2][parL][j][g4];
        v4f p3 = *(const v4f*)&lds_buv[3][parL][j][g4];
        v4f pdt = p0 * p1 * p2 * p3;
        *(v4f*)(msg_out + (size_t)((base >> 2) + parL) * 128 + j * 16 + g4) = pdt;
    }

    // Per-tree LL accumulation to global
    int nslots = 64 >> (2 * level);
    if (nslots < 1) nslots = 1;
    if (tid < 256) {
        int s = tid >> 4, g = tid & 15;
        if (s < nslots) {
            int tree = (base >> (2 * level)) + s;
            atomicAdd(&ll_out[tree * 16 + g], lds_ll[s][g]);
        }
    }
}

// Root (level 0): one node per tree.
__global__ void root_kernel(const float* __restrict__ smB,
                            const float* __restrict__ pc0,
                            const int*   __restrict__ x,
                            const float* __restrict__ msg0,
                            float* __restrict__ ll_out) {
    int tid = blockIdx.x * blockDim.x + threadIdx.x;
    if (tid >= 512) return;
    int tree = tid >> 4, g = tid & 15;
    int xi = x[tree];
    float nu = 0.0f;
    #pragma unroll
    for (int c = 0; c < 8; ++c) {
        float bem  = smB[c * 4096 + xi * 16 + g];
        float pr   = pc0[c * 16 + g];
        float prev = pr * bem;
        nu += prev * prev * msg0[(size_t)tree * 128 + c * 16 + g];
    }
    atomicAdd(&ll_out[tree * 16 + g], fast_log(nu));
}

extern "C" void kernel_launch(void* const* d_in, const int* in_sizes, int n_in,
                              void* d_out, int out_size, void* d_ws, size_t ws_size,
                              hipStream_t stream) {
    const float* A  = (const float*)d_in[0];   // (8,8,4,16)
    const float* B  = (const float*)d_in[1];   // (8,256,16)
    const float* Pi = (const float*)d_in[2];   // (8,16)
    const int*   x  = (const int*)d_in[3];     // (174752,)
    // d_in[4]=pos, d_in[5]=batch: structure derived analytically

    float* out = (float*)d_out;                // (32,16)
    float* ws  = (float*)d_ws;

    float* smA = ws;                     // 4096
    float* smB = ws + 4096;              // 32768
    float* pc  = ws + 4096 + 32768;      // 5461 * 128 = 699008 (all prior levels)
    float* msg = pc + 699008;            // msg levels 0..5: sum = 43680*128 floats

    // Prior-level offsets (nodes): POFF[l] = (4^l - 1)/3
    const int POFF[7] = {0, 1, 5, 21, 85, 341, 1365};
    // x offsets per level
    const int OFFX[7] = {0, 32, 160, 672, 2720, 10912, 43680};
    // msg offsets (floats) for levels 0..5
    size_t MOFF[6];
    {
        size_t acc = 0;
        for (int l = 0; l < 6; ++l) { MOFF[l] = acc; acc += (size_t)32 * (1u << (2 * l)) * 128; }
    }

    zero_out_kernel<<<1, 512, 0, stream>>>(out);
    softmax_kernel<<<3, 256, 0, stream>>>(A, B, Pi, smA, smB, pc /* level-0 prior = sm_Pi */);

    for (int l = 1; l <= 6; ++l) {
        int total = (1 << (2 * l)) * 128;
        prior_kernel<<<(total + 255) / 256, 256, 0, stream>>>(smA, pc, l, POFF[l - 1], POFF[l]);
    }

    for (int l = 6; l >= 1; --l) {
        int sz = 32 << (2 * l);
        up_kernel<<<sz / 64, 128, 0, stream>>>(
            smA, smB, pc, x,
            (l == 6) ? (const float*)nullptr : (msg + MOFF[l]),
            msg + MOFF[l - 1],
            out, l, POFF[l], OFFX[l]);
    }

    root_kernel<<<2, 256, 0, stream>>>(smB, pc, x, msg + MOFF[0], out);
}